// GELU126_39857296507316
// MI455X (gfx1250) — compile-verified
//
#include <hip/hip_runtime.h>
#include <math.h>

// ---------------------------------------------------------------------------
// Gated-EMA GELU, (B,T,D) = (8,2048,1024) f32.
// HBM-bound op (~128MB min traffic -> ~5.5us @ 23.3TB/s). The T-scan is a
// first-order linear recurrence, parallelized by chunking T into C chunks:
//   phase1: per-chunk affine offsets  (parallel over B*D*C)
//   phase2: stitch chunk-entry states (sequential over C, parallel over B*D)
//   phase3: gates + gelu from chunk-entry states (parallel over B*D*C)
// x (64MB) stays resident in the 192MB L2 between phase1 and phase3.
// Phase1/3 stage x through LDS with gfx1250 async-to-LDS DMA (ASYNCcnt),
// double buffered, so the serial EMA chain never waits on global latency.
// ---------------------------------------------------------------------------

namespace {

constexpr int kB   = 8;
constexpr int kT   = 2048;
constexpr int kD   = 1024;
constexpr int kBlk = 256;              // threads per block = 8 wave32
constexpr int kC   = 32;               // chunks along T
constexpr int kL   = kT / kC;          // 64 timesteps per chunk
constexpr int kTS  = 16;               // timesteps per staged slab
constexpr int kNS  = kL / kTS;         // 4 slabs per chunk
constexpr int kBD  = kB * kD;          // 8192 scan channels
constexpr int kDBlks = kD / kBlk;      // 4 d-blocks

#define EPS_F            1e-5f
#define EPS_VAR_F        1e-4f
#define SQRT_2_OVER_PI_F 0.79788456080286535588f

__device__ __forceinline__ float fast_tanh(float v) {
#if __has_builtin(__builtin_amdgcn_tanhf)
  return __builtin_amdgcn_tanhf(v);        // v_tanh_f32 (gfx1250 trans op)
#else
  return tanhf(v);
#endif
}

__device__ __forceinline__ float gelu_tanh_f(float v) {
  float u = SQRT_2_OVER_PI_F * fmaf(0.044715f * v, v * v, v);
  return 0.5f * v * (1.0f + fast_tanh(u));
}

struct Params { float dd, om, alpha, sigma; };

__device__ __forceinline__ Params load_params(const float* logit_decay,
                                              const float* log_alpha,
                                              const float* log_sigma) {
  Params p;
  p.dd    = 1.0f / (1.0f + expf(-logit_decay[0]));   // sigmoid
  p.om    = 1.0f - p.dd;
  p.alpha = log1pf(expf(log_alpha[0]));              // softplus
  p.sigma = log1pf(expf(log_sigma[0]));              // softplus
  return p;
}

// ---- CDNA5 async global->LDS staging (guarded; falls back to plain loads) --
#if defined(__HIP_DEVICE_COMPILE__) && \
    __has_builtin(__builtin_amdgcn_global_load_async_to_lds_b32)
#define ATHENA_ASYNC_LDS 1
#endif

#if defined(ATHENA_ASYNC_LDS)
typedef __attribute__((address_space(1))) int* gptr_t;   // global int*
typedef __attribute__((address_space(3))) int* lptr_t;   // LDS int*

__device__ __forceinline__ void async_copy_b32(const float* g, float* l) {
  // addrspacecast generic->as(1)/as(3); builtin signature (per clang
  // diagnostic): (as1 int*, as3 int*, imm offset, imm cpol)
  __builtin_amdgcn_global_load_async_to_lds_b32(
      (gptr_t)(float*)g, (lptr_t)l, /*offset=*/0, /*cpol=*/0);
}
template <int N>
__device__ __forceinline__ void wait_async_le() {
#if __has_builtin(__builtin_amdgcn_s_wait_asynccnt)
  __builtin_amdgcn_s_wait_asynccnt(N);
#else
  asm volatile("s_wait_asynccnt %0" :: "n"(N) : "memory");
#endif
}
#endif

// Fetch one kTS-deep slab of this thread's column into xr[], using async DMA
// double-buffering when available. gcol points at (t0 of chunk, this d).
// Shared by phase1 and phase3 via macro to keep both kernels identical here.
#define SLAB_LOOP_BEGIN(gcol)                                                  \
  for (int sl = 0; sl < kNS; ++sl) {                                           \
    float xr[kTS];                                                             \
    SLAB_FETCH(gcol, sl)

#define SLAB_LOOP_END() }

#if defined(ATHENA_ASYNC_LDS)
#define SLAB_PROLOGUE(gcol)                                                    \
  __shared__ float tile[2][kTS * kBlk];                                        \
  _Pragma("unroll")                                                            \
  for (int tt = 0; tt < kTS; ++tt)                                             \
    async_copy_b32((gcol) + (size_t)tt * kD, &tile[0][tt * kBlk + threadIdx.x]);

#define SLAB_FETCH(gcol, sl)                                                   \
  {                                                                            \
    int buf = (sl) & 1;                                                        \
    if ((sl) + 1 < kNS) {                                                      \
      int nb = buf ^ 1;                                                        \
      _Pragma("unroll")                                                        \
      for (int tt = 0; tt < kTS; ++tt)                                         \
        async_copy_b32((gcol) + (size_t)(((sl) + 1) * kTS + tt) * kD,          \
                       &tile[nb][tt * kBlk + threadIdx.x]);                    \
      wait_async_le<kTS>();   /* in-order: previous slab fully landed */       \
    } else {                                                                   \
      wait_async_le<0>();                                                      \
    }                                                                          \
    _Pragma("unroll")                                                          \
    for (int tt = 0; tt < kTS; ++tt) xr[tt] = tile[buf][tt * kBlk + threadIdx.x]; \
  }
#else
#define SLAB_PROLOGUE(gcol)
#define SLAB_FETCH(gcol, sl)                                                   \
  _Pragma("unroll")                                                            \
  for (int tt = 0; tt < kTS; ++tt)                                             \
    xr[tt] = (gcol)[(size_t)((sl) * kTS + tt) * kD];
#endif

// --------------------------- phase 1: chunk partials -----------------------
// Chunk c>0:  S_out = d^L * S_in + (bm, bs)   -> store (bm, bs)
// Chunk 0:    absolute state after t = L-1 (init at t=0 is (x0, x0^2))
__global__ __launch_bounds__(kBlk) void phase1_partials(
    const float* __restrict__ x, const float* __restrict__ logit_decay,
    float* __restrict__ part_m, float* __restrict__ part_s) {
  float dd = 1.0f / (1.0f + expf(-logit_decay[0]));
  float om = 1.0f - dd;

  int bid  = blockIdx.x;
  int c    = bid % kC;
  int dblk = (bid / kC) % kDBlks;
  int b    = bid / (kC * kDBlks);
  int d    = dblk * kBlk + threadIdx.x;
  bool first_chunk = (c == 0);

  const float* gcol = x + ((size_t)b * kT + (size_t)c * kL) * kD + d;

  float bm = 0.0f, bs = 0.0f;
  SLAB_PROLOGUE(gcol)
  SLAB_LOOP_BEGIN(gcol)
    #pragma unroll
    for (int tt = 0; tt < kTS; ++tt) {
      float xt = xr[tt];
      if (first_chunk && sl == 0 && tt == 0) {
        bm = xt; bs = xt * xt;                      // carry init (x0, x0^2)
      } else {
        bm = fmaf(dd, bm, om * xt);
        bs = fmaf(dd, bs, om * xt * xt);
      }
    }
  SLAB_LOOP_END()

  size_t o = (size_t)c * kBD + (size_t)b * kD + d;
  part_m[o] = bm;
  part_s[o] = bs;
}

// --------------------------- phase 2: stitch -------------------------------
// Sequential over C=32 chunks, one thread per channel (coalesced, trivial).
__global__ __launch_bounds__(kBlk) void phase2_stitch(
    const float* __restrict__ logit_decay,
    const float* __restrict__ part_m, const float* __restrict__ part_s,
    float* __restrict__ in_m, float* __restrict__ in_s) {
  int ch = blockIdx.x * kBlk + threadIdx.x;        // 0..kBD-1
  float dd = 1.0f / (1.0f + expf(-logit_decay[0]));
  float dL = powf(dd, (float)kL);
  float m = part_m[ch], s = part_s[ch];            // S_end(chunk 0)
  for (int c = 1; c < kC; ++c) {
    size_t o = (size_t)c * kBD + ch;
    in_m[o] = m;                                   // state entering chunk c
    in_s[o] = s;
    m = fmaf(dL, m, part_m[o]);
    s = fmaf(dL, s, part_s[o]);
  }
}

// --------------------------- phase 3: gates + gelu -------------------------
__global__ __launch_bounds__(kBlk) void phase3_apply(
    const float* __restrict__ x, const float* __restrict__ logit_decay,
    const float* __restrict__ log_alpha, const float* __restrict__ log_sigma,
    const float* __restrict__ in_m, const float* __restrict__ in_s,
    float* __restrict__ out) {
  Params p = load_params(logit_decay, log_alpha, log_sigma);

  int bid  = blockIdx.x;
  int c    = bid % kC;
  int dblk = (bid / kC) % kDBlks;
  int b    = bid / (kC * kDBlks);
  int d    = dblk * kBlk + threadIdx.x;
  bool first_chunk = (c == 0);

  size_t colbase = ((size_t)b * kT + (size_t)c * kL) * kD + d;
  const float* gcol = x + colbase;
  float* ocol = out + colbase;

  float m = 0.0f, s2 = 0.0f;
  if (!first_chunk) {
    size_t o = (size_t)c * kBD + (size_t)b * kD + d;
    m = in_m[o]; s2 = in_s[o];
  }

  SLAB_PROLOGUE(gcol)
  SLAB_LOOP_BEGIN(gcol)
    #pragma unroll
    for (int tt = 0; tt < kTS; ++tt) {
      float xt = xr[tt];
      float res;
      if (first_chunk && sl == 0 && tt == 0) {
        res = gelu_tanh_f(xt);                     // gate = 1 at t = 0
        m = xt; s2 = xt * xt;
      } else {
        float var = fmaxf(fmaf(-m, m, s2), EPS_VAR_F);
        float std = __builtin_amdgcn_sqrtf(var);
        float z   = (xt - m) * __builtin_amdgcn_rcpf(std + EPS_F);
        float g   = fmaf(p.alpha, fast_tanh(p.sigma * fabsf(z)), 1.0f);
        res = gelu_tanh_f(xt * g);
        m  = fmaf(p.dd, m,  p.om * xt);
        s2 = fmaf(p.dd, s2, p.om * xt * xt);
      }
      ocol[(size_t)(sl * kTS + tt) * kD] = res;    // coalesced B32 stores
    }
  SLAB_LOOP_END()
}

}  // namespace

extern "C" void kernel_launch(void* const* d_in, const int* in_sizes, int n_in,
                              void* d_out, int out_size, void* d_ws, size_t ws_size,
                              hipStream_t stream) {
  const float* x           = (const float*)d_in[0];
  const float* logit_decay = (const float*)d_in[1];
  const float* log_alpha   = (const float*)d_in[2];
  const float* log_sigma   = (const float*)d_in[3];
  float* out = (float*)d_out;

  // Scratch layout: 4 arrays of kC*kBD f32 (1 MB each, 4 MB total).
  float* part_m = (float*)d_ws;
  float* part_s = part_m + (size_t)kC * kBD;
  float* in_m   = part_s + (size_t)kC * kBD;
  float* in_s   = in_m   + (size_t)kC * kBD;

  dim3 gridHeavy(kC * kDBlks * kB);   // 1024 blocks x 8 wave32 = 8192 waves
  phase1_partials<<<gridHeavy, kBlk, 0, stream>>>(x, logit_decay, part_m, part_s);
  phase2_stitch<<<dim3(kBD / kBlk), kBlk, 0, stream>>>(logit_decay, part_m, part_s,
                                                       in_m, in_s);
  phase3_apply<<<gridHeavy, kBlk, 0, stream>>>(x, logit_decay, log_alpha, log_sigma,
                                               in_m, in_s, out);
}